// LSSOT_loss_61160334295354
// MI455X (gfx1250) — compile-verified
//
#include <hip/hip_runtime.h>
#include <hip/hip_bf16.h>
#include <math.h>

// ---------------------------------------------------------------------------
// LSSOT loss for MI455X (gfx1250).
//  x1,x2: [8,2048,512] f32  ->  out: [8] f32
// Pipeline:
//   1. make_W      : deterministic Gaussian Z[256,512,2] + Gram-Schmidt -> W[512,512]
//   2. gemm_angle  : per-batch (2048x512)x(512x512) GEMM via V_WMMA_F32_16X16X4_F32,
//                    K-chunks staged into double-buffered LDS with
//                    GLOBAL_LOAD_ASYNC_TO_LDS (ASYNCcnt) + fused atan2 epilogue
//   3. sort_rows   : LDS bitonic sort of each (b,l) row of 2048 angles, + mean->alpha
//   4. loss_kernel : circular-CDF inversion (nested binary search over LDS) + reduce
// ---------------------------------------------------------------------------

#define D      512
#define NSAMP  2048
#define LPROJ  256
#define BATCH  8
#define NREF   2000
#define MNEW   6000
#define COLS   (2 * LPROJ)        // 512 projection columns (l,k) flattened
#define RROWS  (BATCH * LPROJ)    // 2048 (b,l) rows

#define KCH    32                 // K-chunk staged per LDS buffer
#define APAD   34                 // A row pitch in LDS (floats): conflict-free fragments

typedef __attribute__((ext_vector_type(2))) float v2f;
typedef __attribute__((ext_vector_type(8))) float v8f;

// ---------------------------------------------------------------------------
// CDNA5 async global->LDS copies (tracked with ASYNCcnt).
// LDS destination address is the low 32 bits of the flat pointer
// (ISA 10.2: LDS aperture maps addr[31:0] directly to the LDS offset).
// ---------------------------------------------------------------------------
__device__ __forceinline__ unsigned lds_off(const void* p) {
  return (unsigned)(uintptr_t)p;
}
__device__ __forceinline__ void async_copy_b64(unsigned lds, const float* gsrc) {
  asm volatile("global_load_async_to_lds_b64 %0, %1, off"
               :: "v"(lds), "v"(gsrc) : "memory");
}
__device__ __forceinline__ void async_copy_b128(unsigned lds, const float* gsrc) {
  asm volatile("global_load_async_to_lds_b128 %0, %1, off"
               :: "v"(lds), "v"(gsrc) : "memory");
}
__device__ __forceinline__ void wait_async0() {
  asm volatile("s_wait_asynccnt 0x0" ::: "memory");
}

// ---------------------------------------------------------------------------
// Deterministic pseudo-random normal (Box-Muller over an integer hash).
// ---------------------------------------------------------------------------
__device__ __forceinline__ unsigned hash_u32(unsigned x) {
  x ^= x >> 16; x *= 0x7feb352dU;
  x ^= x >> 15; x *= 0x846ca68bU;
  x ^= x >> 16;
  return x;
}

__device__ __forceinline__ float gauss_ld(int l, int d, int k) {
  unsigned idx = (unsigned)((l * D + d) * 2 + k);
  unsigned h1 = hash_u32(idx * 2654435761U + 0x9E3779B9U);
  unsigned h2 = hash_u32(idx * 0x85EBCA6BU + 0xC2B2AE35U);
  float u1 = (float)(h1 >> 8) * (1.0f / 16777216.0f) + 1e-7f;
  float u2 = (float)(h2 >> 8) * (1.0f / 16777216.0f);
  return sqrtf(-2.0f * logf(u1)) * __cosf(6.283185307179586f * u2);
}

// Block-wide sum for 256-thread blocks (wave32: 8 waves). Result valid on tid 0.
__device__ __forceinline__ float block_sum_256(float v, float* red) {
  int lane = threadIdx.x & 31, w = threadIdx.x >> 5;
#pragma unroll
  for (int o = 16; o > 0; o >>= 1) v += __shfl_down(v, o, 32);
  if (lane == 0) red[w] = v;
  __syncthreads();
  float r = (threadIdx.x < 8) ? red[threadIdx.x] : 0.0f;
  if (w == 0) {
#pragma unroll
    for (int o = 4; o > 0; o >>= 1) r += __shfl_down(r, o, 32);
  }
  __syncthreads();
  return r;
}

// ---------------------------------------------------------------------------
// Kernel 1: build W[d][c], c = 2*l + k, from Gram-Schmidt of Z[l][:,0:2].
// One 256-thread block per projection l; each thread owns d = {t, t+256}.
// ---------------------------------------------------------------------------
__global__ void make_W(float* __restrict__ W) {
  __shared__ float red[8];
  __shared__ float bc;
  int l = blockIdx.x;
  int t = threadIdx.x;

  float z0a = gauss_ld(l, t, 0),       z1a = gauss_ld(l, t, 1);
  float z0b = gauss_ld(l, t + 256, 0), z1b = gauss_ld(l, t + 256, 1);

  float s = block_sum_256(z0a * z0a + z0b * z0b, red);
  if (t == 0) bc = s;
  __syncthreads();
  float inv0 = rsqrtf(fmaxf(bc, 1e-30f));
  float q0a = z0a * inv0, q0b = z0b * inv0;

  s = block_sum_256(q0a * z1a + q0b * z1b, red);
  if (t == 0) bc = s;
  __syncthreads();
  float dot = bc;
  float wa = z1a - dot * q0a, wb = z1b - dot * q0b;

  s = block_sum_256(wa * wa + wb * wb, red);
  if (t == 0) bc = s;
  __syncthreads();
  float inv1 = rsqrtf(fmaxf(bc, 1e-30f));

  W[(size_t)t * COLS + 2 * l]             = q0a;
  W[(size_t)t * COLS + 2 * l + 1]         = wa * inv1;
  W[(size_t)(t + 256) * COLS + 2 * l]     = q0b;
  W[(size_t)(t + 256) * COLS + 2 * l + 1] = wb * inv1;
}

// ---------------------------------------------------------------------------
// Kernel 2: P[b] = X[b] (2048x512) * W (512x512), fused angle epilogue.
// Block (8 waves) owns 128 M-rows x 16 N-cols; each wave one 16x16 WMMA tile.
// K staged in 32-wide chunks via async global->LDS copies, double buffered:
//   A chunk [128][KCH] pad 34 : bank = (34m+k) % 64  -> conflict-free fragments
//   B chunk [KCH][16]  pad  0 : bank = (16k+n) % 64  -> conflict-free fragments
// WMMA f32 16x16x4: lanes 0-15 hold K={kk,kk+1}, lanes 16-31 K={kk+2,kk+3};
// C: lane holds N=lane&15, rows M = v + 8*(lane>=16). Adjacent columns are
// (p0,p1) of one projection -> shfl_xor(.,1) pairs them for atan2.
// ---------------------------------------------------------------------------
__global__ void __launch_bounds__(256) gemm_angle(const float* __restrict__ X,
                                                  const float* __restrict__ W,
                                                  float* __restrict__ theta) {
  __shared__ float As[2][128 * APAD];
  __shared__ float Bs[2][KCH * 16];

  const int t    = threadIdx.x;
  const int lane = t & 31;
  const int wave = t >> 5;
  const int b    = blockIdx.y;
  const int strip = blockIdx.x >> 5;     // 16 strips of 128 rows
  const int tn    = blockIdx.x & 31;     // 32 N-tiles of 16 cols
  const int m_base = strip * 128;
  const int n_base = tn * 16;

  const float* Xb = X + (size_t)b * NSAMP * D;

  // staging assignment: A: thread t owns row r=t>>1, half h=(t&1)*16 -> 8x b64
  //                     B: threads t<128 own (k=t>>2, q=t&3)        -> 1x b128
  const int ar = t >> 1;
  const int ah = (t & 1) * 16;

  auto issue = [&](int c, int buf) {
    const int kk = c * KCH;
    const float* ga = Xb + (size_t)(m_base + ar) * D + kk + ah;
    unsigned la = lds_off(&As[buf][ar * APAD + ah]);
#pragma unroll
    for (int f = 0; f < 8; ++f)
      async_copy_b64(la + f * 8, ga + f * 2);
    if (t < 128) {
      const int k = t >> 2, q = t & 3;
      async_copy_b128(lds_off(&Bs[buf][k * 16 + q * 4]),
                      W + (size_t)(kk + k) * COLS + n_base + q * 4);
    }
    if (c + 2 < D / KCH)   // keep the next-next A chunk warming toward L2
      __builtin_prefetch(ga + 2 * KCH, 0, 1);
  };

  const int mA   = lane & 15;
  const int kOff = (lane >> 4) << 1;     // 0 for lanes 0-15, 2 for lanes 16-31
  const int nB   = lane & 15;
  const int myM  = wave * 16;            // wave's row slice inside the strip

  v8f acc = {};
  issue(0, 0);
  for (int c = 0; c < D / KCH; ++c) {
    wait_async0();           // this wave's async LDS writes landed
    __syncthreads();         // everyone's chunk-c data visible; chunk c-1 reads done
    if (c + 1 < D / KCH) issue(c + 1, (c + 1) & 1);

    const float* Ab = &As[c & 1][(myM + mA) * APAD + kOff];
    const float* Bb = &Bs[c & 1][nB];
#pragma unroll
    for (int k2 = 0; k2 < KCH; k2 += 4) {
      v2f av, bv;
      av.x = Ab[k2];
      av.y = Ab[k2 + 1];
      bv.x = Bb[(k2 + kOff) * 16];
      bv.y = Bb[(k2 + kOff + 1) * 16];
#if __has_builtin(__builtin_amdgcn_wmma_f32_16x16x4_f32)
      acc = __builtin_amdgcn_wmma_f32_16x16x4_f32(
          /*neg_a=*/false, av, /*neg_b=*/false, bv,
          /*c_mod=*/(short)0, acc, /*reuse_a=*/false, /*reuse_b=*/false);
#else
      acc[0] += av.x * bv.x + av.y * bv.y;   // placeholder, never silent-empty
#endif
    }
  }

  // Epilogue: pair (p0,p1) across adjacent lanes, emit theta[b][l][m].
  int lidx = (n_base + (lane & 15)) >> 1;
  float* thRow = theta + ((size_t)b * LPROJ + lidx) * NSAMP;
  int mOff = m_base + myM + ((lane >> 4) << 3);
#pragma unroll
  for (int v = 0; v < 8; ++v) {
    float self  = acc[v];
    float other = __shfl_xor(self, 1, 32);
    if (!(lane & 1)) {
      float p0 = self, p1 = other;
      float inv_n = rsqrtf(p0 * p0 + p1 * p1 + 1e-30f);
      float a0 = p0 * inv_n, a1 = p1 * inv_n;
      float den = -a0;
      if (fabsf(den) < 1e-10f) den = 1e-10f;
      float th = (atan2f(-a1, den) + 3.14159265358979f) * 0.15915494309189535f;
      thRow[mOff + v] = th;
    }
  }
}

// ---------------------------------------------------------------------------
// Kernel 3: per-row bitonic sort (2048 elems) in LDS + mean -> alpha.
// 1024 threads, each owns indices {t, t+1024}. 66 compare-exchange steps.
// ---------------------------------------------------------------------------
__global__ void __launch_bounds__(1024) sort_rows(float* __restrict__ data,
                                                  float* __restrict__ alpha) {
  __shared__ float s[NSAMP];
  __shared__ float red[32];
  int row = blockIdx.x;
  float* p = data + (size_t)row * NSAMP;
  int t = threadIdx.x;

  float v0 = p[t], v1 = p[t + 1024];
  s[t] = v0; s[t + 1024] = v1;

  float sum = v0 + v1;
  int lane = t & 31, w = t >> 5;
#pragma unroll
  for (int o = 16; o > 0; o >>= 1) sum += __shfl_down(sum, o, 32);
  if (lane == 0) red[w] = sum;
  __syncthreads();
  if (t < 32) {
    float r = red[t];
#pragma unroll
    for (int o = 16; o > 0; o >>= 1) r += __shfl_down(r, o, 32);
    if (t == 0) alpha[row] = r * (1.0f / NSAMP) - 0.5f;
  }

  for (unsigned k = 2; k <= NSAMP; k <<= 1) {
    for (unsigned j = k >> 1; j > 0; j >>= 1) {
      __syncthreads();
#pragma unroll
      for (int half = 0; half < 2; ++half) {
        unsigned i = (unsigned)t + (unsigned)half * 1024u;
        unsigned ixj = i ^ j;
        if (ixj > i) {
          float a = s[i], bb = s[ixj];
          bool up = ((i & k) == 0);
          bool sw = up ? (a > bb) : (a < bb);
          if (sw) { s[i] = bb; s[ixj] = a; }
        }
      }
    }
  }
  __syncthreads();
  p[t] = s[t];
  p[t + 1024] = s[t + 1024];
}

// ---------------------------------------------------------------------------
// Kernel 4: circular-CDF inversion + loss reduction.
//   cdf(x_i) = floor(x_i) + F(frac(x_i)),  F = linear interp of empirical CDF.
//   inv(q)   = grid-based inverse over xnew = linspace(-1,2,6000) (matches
//              the reference's searchsorted-based _interp1d, incl. end extrap).
// ---------------------------------------------------------------------------
__device__ __forceinline__ float cdf_at(const float* s, int i) {
  float x = -1.0f + 3.0f * (float)i * (1.0f / (float)(MNEW - 1));
  float ix = floorf(x);
  float rest = x - ix;
  int lo = 0, hi = NSAMP;
  while (lo < hi) {
    int mid = (lo + hi) >> 1;
    if (s[mid] < rest) lo = mid + 1; else hi = mid;
  }
  int idx = lo - 1;
  idx = idx < 0 ? 0 : (idx > NSAMP - 2 ? NSAMP - 2 : idx);
  float x0 = s[idx], x1 = s[idx + 1];
  float slope = (1.0f / (float)NSAMP) / fmaxf(x1 - x0, 1e-12f);
  float F = (float)(idx + 1) * (1.0f / (float)NSAMP) + slope * (rest - x0);
  return ix + F;
}

__device__ __forceinline__ float invert_cdf(const float* s, float q) {
  int lo = 0, hi = MNEW;
  while (lo < hi) {
    int mid = (lo + hi) >> 1;
    if (cdf_at(s, mid) < q) lo = mid + 1; else hi = mid;
  }
  int idx = lo - 1;
  idx = idx < 0 ? 0 : (idx > MNEW - 2 ? MNEW - 2 : idx);
  float c0 = cdf_at(s, idx), c1 = cdf_at(s, idx + 1);
  float x0   = -1.0f + 3.0f * (float)idx * (1.0f / (float)(MNEW - 1));
  float step = 3.0f * (1.0f / (float)(MNEW - 1));
  return x0 + step * (q - c0) / fmaxf(c1 - c0, 1e-12f);
}

__global__ void __launch_bounds__(256) loss_kernel(const float* __restrict__ s1g,
                                                   const float* __restrict__ s2g,
                                                   const float* __restrict__ a1,
                                                   const float* __restrict__ a2,
                                                   float* __restrict__ out) {
  __shared__ float s1[NSAMP];
  __shared__ float s2[NSAMP];
  __shared__ float red[8];
  int row = blockIdx.x;  // [0, 2048) == (b*256 + l)
  for (int i = threadIdx.x; i < NSAMP; i += blockDim.x) {
    s1[i] = s1g[(size_t)row * NSAMP + i];
    s2[i] = s2g[(size_t)row * NSAMP + i];
  }
  __syncthreads();
  float al1 = a1[row], al2 = a2[row];

  float acc = 0.0f;
  for (int j = threadIdx.x; j < NREF; j += blockDim.x) {
    float r  = (float)j * (1.0f / (float)NREF);
    float e1 = invert_cdf(s1, r - al1) - r;
    float e2 = invert_cdf(s2, r - al2) - r;
    float diff = fabsf(e2 - e1);
    float c = fminf(diff, 1.0f - diff);
    acc += c * c;
  }
  float total = block_sum_256(acc, red);
  if (threadIdx.x == 0)
    atomicAdd(&out[row >> 8], sqrtf(total) * (1.0f / (float)LPROJ));
}

// ---------------------------------------------------------------------------
// Host-side launcher.
// Workspace layout (floats): W[512*512] | th1[8*256*2048] | th2[...] | a1[2048] | a2[2048]
//   total ~34.6 MB.
// ---------------------------------------------------------------------------
extern "C" void kernel_launch(void* const* d_in, const int* in_sizes, int n_in,
                              void* d_out, int out_size, void* d_ws, size_t ws_size,
                              hipStream_t stream) {
  const float* x1 = (const float*)d_in[0];
  const float* x2 = (const float*)d_in[1];
  float* out = (float*)d_out;

  float* W   = (float*)d_ws;
  float* th1 = W + (size_t)D * COLS;
  float* th2 = th1 + (size_t)BATCH * LPROJ * NSAMP;
  float* al1 = th2 + (size_t)BATCH * LPROJ * NSAMP;
  float* al2 = al1 + RROWS;

  hipMemsetAsync(out, 0, (size_t)out_size * sizeof(float), stream);

  make_W<<<LPROJ, 256, 0, stream>>>(W);

  // 16 M-strips x 32 N-tiles = 512 blocks per batch; grid.y = batch
  gemm_angle<<<dim3(512, BATCH), 256, 0, stream>>>(x1, W, th1);
  gemm_angle<<<dim3(512, BATCH), 256, 0, stream>>>(x2, W, th2);

  sort_rows<<<RROWS, 1024, 0, stream>>>(th1, al1);
  sort_rows<<<RROWS, 1024, 0, stream>>>(th2, al2);

  loss_kernel<<<RROWS, 256, 0, stream>>>(th1, th2, al1, al2, out);
}